// EdgeNet_11123965297188
// MI455X (gfx1250) — compile-verified
//
#include <hip/hip_runtime.h>
#include <hip/hip_bf16.h>

// ---------------------------------------------------------------------------
// Problem constants (fixed by reference setup_inputs)
// ---------------------------------------------------------------------------
static constexpr int N_NODES = 4096;
static constexpr int N_EDGES = 8192;
static constexpr int N_FEAT  = 5;     // per-matrix features; B has 10 cols
static constexpr int N_WIRES = 10;
static constexpr int NPAD    = 16;    // padded feature dim for WMMA

typedef __attribute__((ext_vector_type(2))) float v2f;
typedef __attribute__((ext_vector_type(8))) float v8f;

// ---------------------------------------------------------------------------
// Kernel 0: Xpad[k][n] = (n < 5) ? X[k][n] : 0   (4096 x 16)
// Removes all predication from the GEMM inner loop.
// ---------------------------------------------------------------------------
__global__ __launch_bounds__(256) void pad_x_kernel(
    const float* __restrict__ X, float* __restrict__ Xpad)
{
    const int idx = blockIdx.x * 256 + threadIdx.x;   // 0 .. 4096*16-1
    const int k = idx >> 4;
    const int n = idx & 15;
    Xpad[idx] = (n < N_FEAT) ? X[k * N_FEAT + n] : 0.0f;
}

// ---------------------------------------------------------------------------
// Kernel 1: Bmat[e][0..4] = (Ro^T X)[e], Bmat[e][5..9] = (Ri^T X)[e]
// One wave computes a 16(edges) x 16(padded feat) tile via f32 WMMA,
// accumulating K=4096 in steps of 4 with V_WMMA_F32_16X16X4_F32.
//   A (16x4): lane m = L%16, k = 2*(L/16) + reg
//   B (4x16): lane n = L%16, k = 2*(L/16) + reg
//   C (16x16): lane n = L%16, m = reg + 8*(L/16)
// Inner loop: 4 unconditional b32 loads + 1 wmma. No exec-mask games.
// ---------------------------------------------------------------------------
__global__ __launch_bounds__(256) void angles_gemm_kernel(
    const float* __restrict__ Xpad,
    const float* __restrict__ Ri,
    const float* __restrict__ Ro,
    float* __restrict__ Bmat)
{
    const int lane = threadIdx.x & 31;
    const int wave = threadIdx.x >> 5;
    const int tile = blockIdx.x * 8 + wave;   // 0..511 (16 edges each)
    const int e0   = tile * 16;
    const int m16  = lane & 15;
    const int half = lane >> 4;

    const float* __restrict__ R = (blockIdx.y == 0) ? Ro : Ri;
    const int colOff            = (blockIdx.y == 0) ? 0  : N_FEAT;

    // Per-lane base pointers; inner loop only bumps by constant strides.
    const float* __restrict__ aPtr = R + (size_t)(2 * half) * N_EDGES + e0 + m16;
    const float* __restrict__ bPtr = Xpad + (2 * half) * NPAD + m16;

    v8f acc = {};
#pragma unroll 4
    for (int k0 = 0; k0 < N_NODES; k0 += 4) {
        v2f a, b;
        a.x = aPtr[0];
        a.y = aPtr[N_EDGES];
        b.x = bPtr[0];
        b.y = bPtr[NPAD];
        acc = __builtin_amdgcn_wmma_f32_16x16x4_f32(
            /*neg_a=*/false, a, /*neg_b=*/false, b,
            /*c_mod=*/(short)0, acc, /*reuse_a=*/false, /*reuse_b=*/false);
        aPtr += (size_t)4 * N_EDGES;
        bPtr += 4 * NPAD;
    }

    if (m16 < N_FEAT) {
#pragma unroll
        for (int r = 0; r < 8; ++r) {
            const int m = r + 8 * half;
            Bmat[(size_t)(e0 + m) * N_WIRES + colOff + m16] = acc[r];
        }
    }
}

// ---------------------------------------------------------------------------
// Kernel 2: 10-qubit TTN simulation, one wave32 per edge.
// State amplitude idx = r*32 + lane (r = register 0..31).
// Wire w lives at index bit b = 9-w:
//   b >= 5 -> register-index bit (b-5)  : in-lane butterflies
//   b <  5 -> lane-index bit b          : cross-lane shuffles
// ---------------------------------------------------------------------------
__device__ __forceinline__ float shflx(float x, int m) {
    return __shfl_xor(x, m, 32);
}

template <int W>
__device__ __forceinline__ void apply_ry(float (&v)[32], float c, float s, int lane) {
    constexpr int B = 9 - W;
    if constexpr (B >= 5) {
        constexpr int st = 1 << (B - 5);
#pragma unroll
        for (int r = 0; r < 32; ++r) {
            if ((r & st) == 0) {
                const float p0 = v[r];
                const float p1 = v[r + st];
                v[r]      = c * p0 - s * p1;
                v[r + st] = s * p0 + c * p1;
            }
        }
    } else {
        constexpr int m = 1 << B;
        const float sgn = (lane & m) ? s : -s;
#pragma unroll
        for (int r = 0; r < 32; ++r) {
            const float p = shflx(v[r], m);
            v[r] = c * v[r] + sgn * p;
        }
    }
}

template <int C, int T>
__device__ __forceinline__ void apply_cnot(float (&v)[32], int lane) {
    constexpr int BC = 9 - C;
    constexpr int BT = 9 - T;
    if constexpr (BC >= 5 && BT >= 5) {
        // ctrl & tgt both in register bits: pure register permutation
        constexpr int cm = 1 << (BC - 5);
        constexpr int tm = 1 << (BT - 5);
#pragma unroll
        for (int r = 0; r < 32; ++r) {
            if ((r & cm) && !(r & tm)) {
                const float t = v[r];
                v[r]      = v[r + tm];
                v[r + tm] = t;
            }
        }
    } else if constexpr (BC >= 5 && BT < 5) {
        // ctrl in register bits, tgt in lane bits: shuffle selected registers
        constexpr int cm    = 1 << (BC - 5);
        constexpr int tmask = 1 << BT;
#pragma unroll
        for (int r = 0; r < 32; ++r) {
            if (r & cm) v[r] = shflx(v[r], tmask);
        }
    } else if constexpr (BC < 5 && BT >= 5) {
        // ctrl in lane bits, tgt in register bits: predicated register swap
        constexpr int cmask = 1 << BC;
        constexpr int tm    = 1 << (BT - 5);
        const bool cb = (lane & cmask) != 0;
#pragma unroll
        for (int r = 0; r < 32; ++r) {
            if ((r & tm) == 0) {
                const float a = v[r];
                const float b = v[r + tm];
                v[r]      = cb ? b : a;
                v[r + tm] = cb ? a : b;
            }
        }
    } else {
        // ctrl & tgt both in lane bits: predicated shuffle
        constexpr int cmask = 1 << BC;
        constexpr int tmask = 1 << BT;
        const bool cb = (lane & cmask) != 0;
#pragma unroll
        for (int r = 0; r < 32; ++r) {
            const float t = shflx(v[r], tmask);
            v[r] = cb ? t : v[r];
        }
    }
}

__global__ __launch_bounds__(256) void ttn_sim_kernel(
    const float* __restrict__ Bmat,
    const float* __restrict__ theta,
    float* __restrict__ out)
{
    const int lane = threadIdx.x & 31;
    const int wave = threadIdx.x >> 5;
    const int e    = blockIdx.x * 8 + wave;   // 8192 edges, 1024 blocks x 8 waves

    // Per-edge embedding angles (wave-uniform loads; identical in all lanes)
    float ac[10], as[10];
#pragma unroll
    for (int w = 0; w < N_WIRES; ++w) {
        const float h = 0.5f * Bmat[(size_t)e * N_WIRES + w];
        sincosf(h, &as[w], &ac[w]);
    }
    // Shared circuit parameters
    float tc[19], ts[19];
#pragma unroll
    for (int j = 0; j < 19; ++j) {
        const float h = 0.5f * theta[j];
        sincosf(h, &ts[j], &tc[j]);
    }

    // |0...0>
    float v[32];
#pragma unroll
    for (int r = 0; r < 32; ++r) v[r] = 0.0f;
    if (lane == 0) v[0] = 1.0f;

    // Embedding layer
    apply_ry<0>(v, ac[0], as[0], lane);
    apply_ry<1>(v, ac[1], as[1], lane);
    apply_ry<2>(v, ac[2], as[2], lane);
    apply_ry<3>(v, ac[3], as[3], lane);
    apply_ry<4>(v, ac[4], as[4], lane);
    apply_ry<5>(v, ac[5], as[5], lane);
    apply_ry<6>(v, ac[6], as[6], lane);
    apply_ry<7>(v, ac[7], as[7], lane);
    apply_ry<8>(v, ac[8], as[8], lane);
    apply_ry<9>(v, ac[9], as[9], lane);
    // TTN layer 1
    apply_ry<0>(v, tc[0], ts[0], lane); apply_ry<1>(v, tc[1], ts[1], lane); apply_cnot<0, 1>(v, lane);
    apply_ry<2>(v, tc[2], ts[2], lane); apply_ry<3>(v, tc[3], ts[3], lane); apply_cnot<3, 2>(v, lane);
    apply_ry<4>(v, tc[4], ts[4], lane); apply_ry<5>(v, tc[5], ts[5], lane); apply_cnot<5, 4>(v, lane);
    apply_ry<6>(v, tc[6], ts[6], lane); apply_ry<7>(v, tc[7], ts[7], lane); apply_cnot<6, 7>(v, lane);
    apply_ry<8>(v, tc[8], ts[8], lane); apply_ry<9>(v, tc[9], ts[9], lane); apply_cnot<8, 9>(v, lane);
    // TTN layer 2
    apply_ry<1>(v, tc[10], ts[10], lane); apply_ry<2>(v, tc[11], ts[11], lane); apply_cnot<1, 2>(v, lane);
    apply_ry<7>(v, tc[12], ts[12], lane); apply_ry<9>(v, tc[13], ts[13], lane); apply_cnot<9, 7>(v, lane);
    // TTN layer 3
    apply_ry<2>(v, tc[14], ts[14], lane); apply_ry<4>(v, tc[15], ts[15], lane); apply_cnot<2, 4>(v, lane);
    // TTN layer 4
    apply_ry<4>(v, tc[16], ts[16], lane); apply_ry<7>(v, tc[17], ts[17], lane); apply_cnot<4, 7>(v, lane);
    apply_ry<7>(v, tc[18], ts[18], lane);

    // Output: P(q7 = 1). Wire 7 -> index bit 2 (a lane bit).
    float loc = 0.0f;
#pragma unroll
    for (int r = 0; r < 32; ++r) loc += v[r] * v[r];
    loc = (lane & 4) ? loc : 0.0f;
#pragma unroll
    for (int off = 16; off >= 1; off >>= 1) loc += __shfl_xor(loc, off, 32);
    if (lane == 0) out[e] = loc;
}

// ---------------------------------------------------------------------------
// Launch
// ---------------------------------------------------------------------------
extern "C" void kernel_launch(void* const* d_in, const int* in_sizes, int n_in,
                              void* d_out, int out_size, void* d_ws, size_t ws_size,
                              hipStream_t stream)
{
    (void)in_sizes; (void)n_in; (void)out_size; (void)ws_size;
    const float* X     = (const float*)d_in[0];   // (4096, 5)
    const float* Ri    = (const float*)d_in[1];   // (4096, 8192)
    const float* Ro    = (const float*)d_in[2];   // (4096, 8192)
    const float* theta = (const float*)d_in[3];   // (19,)
    float* out  = (float*)d_out;                  // (8192,)

    // Workspace layout: Bmat (8192x10 f32 = 327,680 B) then Xpad (4096x16 f32).
    float* Bmat = (float*)d_ws;
    float* Xpad = (float*)((char*)d_ws + (size_t)N_EDGES * N_WIRES * sizeof(float));

    // Zero-pad X to 4096x16 so GEMM B-fragment loads are unconditional.
    pad_x_kernel<<<(N_NODES * NPAD) / 256, 256, 0, stream>>>(X, Xpad);

    // 512 edge-tiles x {Ro, Ri}; 8 waves (tiles) per 256-thread block.
    dim3 gGemm(512 / 8, 2);
    angles_gemm_kernel<<<gGemm, 256, 0, stream>>>(Xpad, Ri, Ro, Bmat);

    // One wave32 per edge; 8 edges per block.
    ttn_sim_kernel<<<N_EDGES / 8, 256, 0, stream>>>(Bmat, theta, out);
}